// ContrastiveLoss_5815385719271
// MI455X (gfx1250) — compile-verified
//
#include <hip/hip_runtime.h>
#include <stdint.h>

typedef __attribute__((ext_vector_type(2))) float v2f;
typedef __attribute__((ext_vector_type(8))) float v8f;

#define N_PTS   8192
#define DIM     256
#define MARGIN  0.3f
#define BM      64      // block rows
#define BN      128     // block cols
#define KC      64      // K chunk staged in LDS
#define LDST    68      // LDS row stride (floats): 272B = 16B aligned, +4 bank rotation/row

__global__ void closs_zero(float* out) { out[0] = 0.0f; }

// CDNA5 async copy global -> LDS (ASYNCcnt-tracked, no VGPR staging).
__device__ __forceinline__ void async_cp_b128(const float* g, float* l) {
    uint32_t lds = (uint32_t)(uintptr_t)l;   // low 32 bits of generic shared ptr = LDS offset
    asm volatile("global_load_async_to_lds_b128 %0, %1, off"
                 :: "v"(lds), "v"(g) : "memory");
}

__device__ __forceinline__ void wait_async0() {
#if __has_builtin(__builtin_amdgcn_s_wait_asynccnt)
    __builtin_amdgcn_s_wait_asynccnt(0);
#else
    asm volatile("s_wait_asynccnt 0" ::: "memory");
#endif
}

__global__ __launch_bounds__(256) void closs_wmma_f32(
    const float* __restrict__ X, const int* __restrict__ tgt, float* __restrict__ out)
{
    __shared__ float As[2][BM * LDST];   // double-buffered A tile
    __shared__ float Bs[2][BN * LDST];   // double-buffered B tile
    __shared__ int   tRow[BM];
    __shared__ int   tCol[BN];
    __shared__ float wsum[8];

    const int tid  = threadIdx.x;
    const int lane = tid & 31;
    const int wid  = tid >> 5;        // 8 waves
    const int wm   = wid & 1;         // wave row 0..1 -> rows wm*32 .. +31
    const int wn   = wid >> 1;        // wave col 0..3 -> cols wn*32 .. +31
    const int rowBase = blockIdx.x * BM;
    const int colBase = blockIdx.y * BN;

    if (tid < BM)                 tRow[tid]      = tgt[rowBase + tid];
    else if (tid < BM + BN)       tCol[tid - BM] = tgt[colBase + (tid - BM)];

    const int r  = tid >> 4;          // staging row 0..15
    const int c4 = (tid & 15) * 4;    // staging col 0..60

    // Issue one K-chunk of async copies into buffer `buf`.
    auto issue = [&](int buf, int kc) {
        #pragma unroll
        for (int p = 0; p < 4; ++p) {          // A: 64 rows x 64 floats
            const int row = r + p * 16;
            async_cp_b128(X + (size_t)(rowBase + row) * DIM + kc + c4,
                          &As[buf][row * LDST + c4]);
        }
        #pragma unroll
        for (int p = 0; p < 8; ++p) {          // B: 128 rows x 64 floats
            const int row = r + p * 16;
            async_cp_b128(X + (size_t)(colBase + row) * DIM + kc + c4,
                          &Bs[buf][row * LDST + c4]);
        }
    };

    // Fragment addressing (ISA 7.12.2, 16x16x4 f32): lanes 0-15 = K{0,1}, 16-31 = K{2,3}
    const int fr = lane & 15;
    const int fk = (lane >> 4) * 2;

    v8f acc00 = {}, acc01 = {}, acc10 = {}, acc11 = {};

    issue(0, 0);
    wait_async0();
    __syncthreads();

    for (int t = 0; t < DIM / KC; ++t) {
        if (t < DIM / KC - 1) issue((t + 1) & 1, (t + 1) * KC);   // prefetch next chunk

        const float* a = &As[t & 1][0];
        const float* b = &Bs[t & 1][0];
        #pragma unroll
        for (int k = 0; k < KC; k += 4) {
            v2f aF0 = *(const v2f*)(a + (wm * 32      + fr) * LDST + k + fk);
            v2f aF1 = *(const v2f*)(a + (wm * 32 + 16 + fr) * LDST + k + fk);
            v2f bF0 = *(const v2f*)(b + (wn * 32      + fr) * LDST + k + fk);
            v2f bF1 = *(const v2f*)(b + (wn * 32 + 16 + fr) * LDST + k + fk);
            acc00 = __builtin_amdgcn_wmma_f32_16x16x4_f32(false, aF0, false, bF0, (short)0, acc00, false, false);
            acc01 = __builtin_amdgcn_wmma_f32_16x16x4_f32(false, aF0, false, bF1, (short)0, acc01, false, false);
            acc10 = __builtin_amdgcn_wmma_f32_16x16x4_f32(false, aF1, false, bF0, (short)0, acc10, false, false);
            acc11 = __builtin_amdgcn_wmma_f32_16x16x4_f32(false, aF1, false, bF1, (short)0, acc11, false, false);
        }

        wait_async0();     // my prefetch for the next buffer has landed
        __syncthreads();   // everyone's landed; everyone done reading current buffer
    }

    // Fused masked-contrastive epilogue.
    // C/D layout: N = lane&15; M = v + 8*(lane>=16).
    const int mAdd = (lane >> 4) * 8;
    const int nIdx = lane & 15;
    const int tcA  = tCol[wn * 32      + nIdx];
    const int tcB  = tCol[wn * 32 + 16 + nIdx];

    float part = 0.0f;
    #pragma unroll
    for (int v = 0; v < 8; ++v) {
        const int trA = tRow[wm * 32      + v + mAdd];
        const int trB = tRow[wm * 32 + 16 + v + mAdd];
        const float s00 = acc00[v], s01 = acc01[v], s10 = acc10[v], s11 = acc11[v];
        part += (trA == tcA) ? ((s00 < 1.0f) ? (1.0f - s00) : 0.0f) : ((s00 > MARGIN) ? s00 : 0.0f);
        part += (trA == tcB) ? ((s01 < 1.0f) ? (1.0f - s01) : 0.0f) : ((s01 > MARGIN) ? s01 : 0.0f);
        part += (trB == tcA) ? ((s10 < 1.0f) ? (1.0f - s10) : 0.0f) : ((s10 > MARGIN) ? s10 : 0.0f);
        part += (trB == tcB) ? ((s11 < 1.0f) ? (1.0f - s11) : 0.0f) : ((s11 > MARGIN) ? s11 : 0.0f);
    }

    // wave32 reduction, then block reduction, then one atomic per block.
    #pragma unroll
    for (int off = 16; off > 0; off >>= 1)
        part += __shfl_down(part, off, 32);
    if (lane == 0) wsum[wid] = part;
    __syncthreads();
    if (tid == 0) {
        float s = 0.0f;
        #pragma unroll
        for (int w = 0; w < 8; ++w) s += wsum[w];
        atomicAdd(out, s * (1.0f / (float)N_PTS));
    }
}

extern "C" void kernel_launch(void* const* d_in, const int* in_sizes, int n_in,
                              void* d_out, int out_size, void* d_ws, size_t ws_size,
                              hipStream_t stream) {
    const float* X   = (const float*)d_in[0];
    const int*   tgt = (const int*)d_in[1];
    float*       out = (float*)d_out;

    closs_zero<<<dim3(1), dim3(1), 0, stream>>>(out);
    dim3 grid(N_PTS / BM, N_PTS / BN);
    closs_wmma_f32<<<grid, dim3(256), 0, stream>>>(X, tgt, out);
}